// RoiCut_8358006358564
// MI455X (gfx1250) — compile-verified
//
#include <hip/hip_runtime.h>
#include <stdint.h>

// RoiCut on MI455X (gfx1250): pure data movement (0 FLOPs, ~1 GB traffic,
// ~44 us roofline at 23.3 TB/s). Strategy:
//   * TDM (tensor_load_to_lds) DMAs each strided [CCHUNK,32,32] fp32 tile
//     (dim0_stride=W, dim1_stride=H*W, 4B elements, rows only 4B-aligned)
//     from HBM into LDS -- exactly the access pattern the Tensor Data Mover
//     exists for; tracked with TENSORcnt, no VALU/VGPR pressure.
//   * Output is contiguous per block, so 256 threads drain LDS with
//     ds_load_b128 and NON-TEMPORAL global_store_b128 (write-once data must
//     not evict the feature map from the 192 MB L2, which has ~1.6x reuse).

#define BOX     32
#define C_      256
#define H_      200
#define W_      200
#define N_      512
#define CCHUNK  16                              // channels per workgroup
#define TILE_FLOATS (CCHUNK * BOX * BOX)        // 16384 floats
#define TILE_BYTES  (TILE_FLOATS * 4)           // 64 KB LDS tile

typedef __attribute__((ext_vector_type(4))) float    v4f;
typedef __attribute__((ext_vector_type(4))) uint32_t v4u;
typedef __attribute__((ext_vector_type(8))) int32_t  v8i;
typedef __attribute__((ext_vector_type(4))) int32_t  v4i;

__device__ __forceinline__ int clampi(int v, int lo, int hi) {
    return v < lo ? lo : (v > hi ? hi : v);
}

__global__ __launch_bounds__(256) void roicut_tdm_kernel(
    const float* __restrict__ fm,     // [8,256,200,200]
    const int*   __restrict__ boxes,  // [512,2] (y0,x0)
    const int*   __restrict__ assoc,  // [512]
    float*       __restrict__ out)    // [512,256,32,32]
{
    extern __shared__ float smem[];

    const int blk  = blockIdx.x;
    const int box  = blk >> 4;              // / (C_/CCHUNK) == /16
    const int cblk = blk & 15;
    const int c0   = cblk * CCHUNK;

    // Block-uniform box metadata, clipped exactly like the reference.
    const int a  = assoc[box];
    int y0 = boxes[2 * box + 0];
    int x0 = boxes[2 * box + 1];
    y0 = clampi(y0, 0, H_ - BOX);
    x0 = clampi(x0, 0, W_ - BOX);

    if (threadIdx.x < 32) {                 // wave 0 issues the TDM DMA
        const uint64_t gaddr = (uint64_t)(uintptr_t)fm +
            4ull * ((((uint64_t)a * C_ + (uint64_t)c0) * H_ +
                     (uint64_t)y0) * W_ + (uint64_t)x0);
        const uint32_t lds = (uint32_t)(uintptr_t)smem;  // low 32 bits = LDS byte offset

        // ---- D# group 0 (128b): count=1, lds_addr, 57b global_addr, type=2
        v4u g0;
        g0.x = 1u;                                             // count=1 (valid), no gather
        g0.y = lds;                                            // [63:32]  lds_addr
        g0.z = (uint32_t)(gaddr & 0xFFFFFFFFull);              // [95:64]  global_addr[31:0]
        g0.w = (uint32_t)((gaddr >> 32) & 0x01FFFFFFull)       // [120:96] global_addr[56:32]
             | (2u << 30);                                     // [127:126] type=2 ("image")

        // ---- D# group 1 (256b)
        // [17:16] data_size=2 (4B); wg_mask=0; no barrier/iterate/pad.
        // tensor_dim0=32 @[79:48], tensor_dim1=32 @[111:80], tile_dim0=32 @[127:112]
        // tile_dim1=32 @[143:128], tile_dim2=CCHUNK @[159:144]
        // tensor_dim0_stride=W @[207:160], tensor_dim1_stride=H*W @[255:208]
        const uint64_t q0 = (uint64_t)0x00020000u
                          | ((uint64_t)(BOX & 0xFFFF) << 48);          // dim0[15:0]
        const uint64_t q1 = (uint64_t)((uint32_t)BOX >> 16)            // dim0[31:16] (=0)
                          | ((uint64_t)BOX << 16)                      // tensor_dim1
                          | ((uint64_t)BOX << 48);                     // tile_dim0
        const uint64_t q2 = (uint64_t)BOX                              // tile_dim1
                          | ((uint64_t)CCHUNK << 16)                   // tile_dim2
                          | ((uint64_t)(uint32_t)W_ << 32);            // dim0_stride[31:0]
        const uint64_t q3 = 0ull                                       // dim0_stride[47:32]
                          | ((uint64_t)(H_ * W_) << 16);               // dim1_stride
        v8i g1;
        g1[0] = (int32_t)(uint32_t)q0; g1[1] = (int32_t)(uint32_t)(q0 >> 32);
        g1[2] = (int32_t)(uint32_t)q1; g1[3] = (int32_t)(uint32_t)(q1 >> 32);
        g1[4] = (int32_t)(uint32_t)q2; g1[5] = (int32_t)(uint32_t)(q2 >> 32);
        g1[6] = (int32_t)(uint32_t)q3; g1[7] = (int32_t)(uint32_t)(q3 >> 32);

        // ---- D# group 2 (128b): tensor_dim2=CCHUNK; dim3 unused
        v4i g2; g2[0] = CCHUNK; g2[1] = 1; g2[2] = 0; g2[3] = 0;
        // ---- D# group 3 (128b): dims 3/4 unused
        v4i g3; g3[0] = 0; g3[1] = 0; g3[2] = 0; g3[3] = 0;

        // Slot 4 on this toolchain (clang-23 / therock-10.0 headers) is an
        // extra int32x8 -- gather-mode index block; zero is neutral since
        // gather_mode=0 in group 0.
        v8i gx; gx[0]=0; gx[1]=0; gx[2]=0; gx[3]=0; gx[4]=0; gx[5]=0; gx[6]=0; gx[7]=0;

        __builtin_amdgcn_tensor_load_to_lds(g0, g1, g2, g3, gx, /*cpol=*/0);
        __builtin_amdgcn_s_wait_tensorcnt(0);   // tile fully in LDS for this wave
    }
    __syncthreads();                            // publish LDS tile to all 8 waves

    // LDS tile is laid out [cc][y][x] contiguously (TDM fill order) --
    // identical to the contiguous destination layout. Drain with b128.
    const v4f* __restrict__ s4 = (const v4f*)smem;
    v4f* __restrict__ o4 =
        (v4f*)(out + ((uint64_t)box * C_ + (uint64_t)c0) * (BOX * BOX));

    constexpr int N4 = TILE_FLOATS / 4;         // 4096 float4 per block
    #pragma unroll
    for (int j = 0; j < N4 / 256; ++j) {
        const int idx = (int)threadIdx.x + j * 256;
        v4f v = s4[idx];                        // ds_load_b128
        __builtin_nontemporal_store(v, &o4[idx]);  // global_store_b128, TH=NT
    }
}

extern "C" void kernel_launch(void* const* d_in, const int* in_sizes, int n_in,
                              void* d_out, int out_size, void* d_ws, size_t ws_size,
                              hipStream_t stream) {
    const float* fm    = (const float*)d_in[0];   // feature_map fp32 [8,256,200,200]
    const int*   boxes = (const int*)d_in[1];     // boxes_yx int32 [512,2]
    const int*   assoc = (const int*)d_in[2];     // assoc int32 [512]
    float*       out   = (float*)d_out;           // [512,256,32,32] fp32

    const dim3 grid(N_ * (C_ / CCHUNK));          // 8192 blocks
    const dim3 block(256);
    roicut_tdm_kernel<<<grid, block, TILE_BYTES, stream>>>(fm, boxes, assoc, out);
}